// MyPipeline_3564822856328
// MI455X (gfx1250) — compile-verified
//
#include <hip/hip_runtime.h>
#include <math.h>

// ---------------- problem constants ----------------
#define NFFT     1024
#define HOPSZ    512
#define NMELS    128
#define NFREQ    513
#define FPAD     528            // 513 padded to 33*16 for WMMA tiling
#define T_IN     1323000
#define NB       16
#define NFRAMES  2584           // 1 + T_IN/HOPSZ
#define MTILES   162            // ceil(NFRAMES/16)
#define E_ELEMS  (NB*NMELS*NFRAMES)   // 5292032 elements per spectrogram

// ---------------- workspace layout (float offsets) ----------------
// DFT table is interleaved: [k/2][f][4] = {cos(k,f), cos(k+1,f), sin(k,f), sin(k+1,f)}
// Mel fb is interleaved:    [k/2][m][2] = {fb(k,m), fb(k+1,m)}
#define WIN_OFF  0
#define DFT_OFF  1024
#define DFT_SZ   ((NFFT/2)*FPAD*4)          // 1081344
#define FB_OFF   (DFT_OFF + DFT_SZ)         // 1082368
#define FB_SZ    ((FPAD/2)*NMELS*2)         // 67584
#define MAX_OFF  (FB_OFF + FB_SZ)           // 1149952  (3 uint max cells)
#define MEL_OFF  (MAX_OFF + 8)
#define HARM_OFF (MEL_OFF + E_ELEMS)
#define PERC_OFF (HARM_OFF + E_ELEMS)
// total ~17.03M floats = ~68.1 MB of d_ws

typedef float v2f __attribute__((ext_vector_type(2)));
typedef float v4f __attribute__((ext_vector_type(4)));
typedef float v8f __attribute__((ext_vector_type(8)));

__device__ __forceinline__ v8f wmma4(v2f a, v2f b, v8f c) {
  // V_WMMA_F32_16X16X4_F32 : D = A(16x4) * B(4x16) + C(16x16), all f32
  return __builtin_amdgcn_wmma_f32_16x16x4_f32(false, a, false, b, (short)0, c,
                                               false, false);
}

// ================= kernel 1: constant tables =================
__global__ __launch_bounds__(256) void build_tables(float* __restrict__ ws) {
  int gid = blockIdx.x * 256 + threadIdx.x;
  if (gid == 0) {
    unsigned* mx = (unsigned*)(ws + MAX_OFF);
    mx[0] = 0u; mx[1] = 0u; mx[2] = 0u;
  }
  const float TWO_PI = 6.28318530717958647692f;
  if (gid < DFT_OFF) {                       // Hann window (periodic)
    int k = gid;
    ws[WIN_OFF + k] = 0.5f * (1.0f - cosf(TWO_PI * (float)k / 1024.0f));
  } else if (gid < FB_OFF) {                 // interleaved cos/sin DFT table
    int i = gid - DFT_OFF;
    int entry = i & 3;                       // 0,1: cos ; 2,3: sin
    int rest  = i >> 2;
    int f  = rest % FPAD;
    int k2 = rest / FPAD;
    int k  = 2 * k2 + (entry & 1);
    int r  = (k * f) & (NFFT - 1);           // exact phase reduction
    float arg = (float)r * (TWO_PI / 1024.0f);
    ws[gid] = (entry < 2) ? cosf(arg) : sinf(arg);
  } else if (gid < MAX_OFF) {                // interleaved HTK mel fb [k/2][m][2]
    int i = gid - FB_OFF;
    int entry = i & 1;
    int rest  = i >> 1;
    int m  = rest % NMELS;
    int k2 = rest / NMELS;
    int f  = 2 * k2 + entry;                 // frequency bin index
    float v = 0.0f;
    if (f < NFREQ) {
      float freq   = (float)f * (22050.0f / 512.0f);
      float melmax = 2595.0f * log10f(1.0f + 22050.0f / 700.0f);
      float step   = melmax / 129.0f;        // 130 points -> 129 intervals
      float p0 = 700.0f * (exp10f(((float)(m    ) * step) / 2595.0f) - 1.0f);
      float p1 = 700.0f * (exp10f(((float)(m + 1) * step) / 2595.0f) - 1.0f);
      float p2 = 700.0f * (exp10f(((float)(m + 2) * step) / 2595.0f) - 1.0f);
      float down = (freq - p0) / (p1 - p0);
      float up   = (p2 - freq) / (p2 - p1);
      float t = fmaxf(0.0f, fminf(down, up));
      v = t * 2.0f / (p2 - p0);
    }
    ws[gid] = v;                              // rows f>=513 are zero
  }
}

// ================= kernel 2: STFT-power + mel via WMMA =================
// One block: 16 frames of one batch. 256 threads = 8 wave32s.
#define KCHUNK 256
__global__ __launch_bounds__(256) void mel_gemm(const float* __restrict__ wave,
                                                const float* __restrict__ ws,
                                                float* __restrict__ mel) {
  __shared__ float AL[16][KCHUNK + 4];   // windowed frame chunk (padded rows)
  __shared__ float PL[16][FPAD + 4];     // power tile 16 x 528

  const float* __restrict__ win  = ws + WIN_OFF;
  const float* __restrict__ dftT = ws + DFT_OFF;
  const float* __restrict__ fb2  = ws + FB_OFF;

  int b    = blockIdx.x / MTILES;
  int tile = blockIdx.x % MTILES;
  int t0   = tile * 16;
  int tid  = threadIdx.x;
  int lane = tid & 31;
  int wv   = tid >> 5;         // wave id 0..7
  int h    = lane >> 4;        // lane half
  int l16  = lane & 15;

  const float* wb = wave + (long)b * T_IN;

  v8f accC[5], accS[5];
  #pragma unroll
  for (int i = 0; i < 5; ++i)
    for (int j = 0; j < 8; ++j) { accC[i][j] = 0.0f; accS[i][j] = 0.0f; }

  for (int kc = 0; kc < NFFT / KCHUNK; ++kc) {
    __syncthreads();
    // build windowed, reflect-padded frame chunk in LDS
    for (int idx = tid; idx < 16 * KCHUNK; idx += 256) {
      int m  = idx >> 8;
      int kl = idx & (KCHUNK - 1);
      int t  = t0 + m;
      float v = 0.0f;
      if (t < NFRAMES) {
        int k = kc * KCHUNK + kl;
        int o = t * HOPSZ + k - NFFT / 2;          // 'reflect' padding
        o = (o < 0) ? -o : o;
        o = (o >= T_IN) ? (2 * (T_IN - 1) - o) : o;
        v = wb[o] * win[k];
      }
      AL[m][kl] = v;
    }
    __syncthreads();

    #pragma unroll
    for (int i = 0; i < 5; ++i) {
      int ft = wv + 8 * i;                         // F-tile (wave-uniform)
      if (ft < FPAD / 16) {
        int f0 = ft * 16;
        // per-lane column base in the interleaved pair table
        const float* dB = dftT + (long)(f0 + l16) * 4;
        __builtin_prefetch(dB + (long)(kc * (KCHUNK / 2)) * FPAD * 4, 0, 1);
        for (int k = 0; k < KCHUNK; k += 4) {
          int kk  = k + 2 * h;                     // A/B: K = v + 2*(lane>>4)
          v2f a   = *(const v2f*)&AL[l16][kk];
          int k2g = kc * (KCHUNK / 2) + (kk >> 1); // global K pair index
          // one b128: {cos k, cos k+1, sin k, sin k+1} for this column
          v4f q = *(const v4f*)(dB + (long)k2g * (FPAD * 4));
          v2f bc; bc[0] = q[0]; bc[1] = q[1];
          v2f bs; bs[0] = q[2]; bs[1] = q[3];
          accC[i] = wmma4(a, bc, accC[i]);         // Re accumulation
          accS[i] = wmma4(a, bs, accS[i]);         // Im accumulation (sign^2)
        }
      }
    }
  }

  __syncthreads();
  // power = Re^2 + Im^2 into LDS tile (C/D layout: M = v + 8h, N = l16)
  #pragma unroll
  for (int i = 0; i < 5; ++i) {
    int ft = wv + 8 * i;
    if (ft < FPAD / 16) {
      int f0 = ft * 16;
      #pragma unroll
      for (int v = 0; v < 8; ++v) {
        int mr = v + 8 * h;
        float re = accC[i][v], im = accS[i][v];
        PL[mr][f0 + l16] = re * re + im * im;
      }
    }
  }
  __syncthreads();

  // stage 2: mel projection, one 16-wide mel tile per wave (8*16 = 128)
  int n0 = wv * 16;
  const float* fB = fb2 + (long)(n0 + l16) * 2;
  v8f acc;
  #pragma unroll
  for (int j = 0; j < 8; ++j) acc[j] = 0.0f;
  for (int k = 0; k < FPAD; k += 4) {
    int kk = k + 2 * h;
    v2f a  = *(const v2f*)&PL[l16][kk];
    // one b64: {fb(k,m), fb(k+1,m)} for this mel column
    v2f bb = *(const v2f*)(fB + (long)(kk >> 1) * (NMELS * 2));
    acc = wmma4(a, bb, acc);
  }
  int n = n0 + l16;
  #pragma unroll
  for (int v = 0; v < 8; ++v) {
    int t = t0 + v + 8 * h;
    if (t < NFRAMES) mel[(b * NMELS + n) * NFRAMES + t] = acc[v];
  }
}

// ============ kernels 3/4: 31-tap medians (branchless selection) ============
__device__ __forceinline__ float median31(float v[31]) {
  #pragma unroll
  for (int i = 0; i < 16; ++i) {
    #pragma unroll
    for (int j = i + 1; j < 31; ++j) {
      float lo = fminf(v[i], v[j]);
      float hi = fmaxf(v[i], v[j]);
      v[i] = lo; v[j] = hi;
    }
  }
  return v[15];
}

__global__ __launch_bounds__(256) void median_time(const float* __restrict__ mel,
                                                   float* __restrict__ harm) {
  int gid = blockIdx.x * 256 + threadIdx.x;
  if (gid >= E_ELEMS) return;
  int t   = gid % NFRAMES;
  int row = gid / NFRAMES;
  const float* src = mel + row * NFRAMES;
  float v[31];
  #pragma unroll
  for (int d = 0; d < 31; ++d) {                 // 'symmetric' padding
    int j = t + d - 15;
    j = (j < 0) ? (-j - 1) : j;
    j = (j >= NFRAMES) ? (2 * NFRAMES - 1 - j) : j;
    v[d] = src[j];
  }
  harm[gid] = median31(v);
}

__global__ __launch_bounds__(256) void median_mel(const float* __restrict__ mel,
                                                  float* __restrict__ perc) {
  int gid = blockIdx.x * 256 + threadIdx.x;
  if (gid >= E_ELEMS) return;
  int t = gid % NFRAMES;
  int n = (gid / NFRAMES) % NMELS;
  int b = gid / (NFRAMES * NMELS);
  const float* base = mel + (b * NMELS) * NFRAMES + t;
  float v[31];
  #pragma unroll
  for (int d = 0; d < 31; ++d) {
    int j = n + d - 15;
    j = (j < 0) ? (-j - 1) : j;
    j = (j >= NMELS) ? (2 * NMELS - 1 - j) : j;
    v[d] = base[j * NFRAMES];
  }
  perc[gid] = median31(v);
}

// ============ kernel 5: HPSS masks + global maxima (atomic, order-free) ====
__global__ __launch_bounds__(256) void hpss_mask(const float* __restrict__ mel,
                                                 float* __restrict__ harm,
                                                 float* __restrict__ perc,
                                                 unsigned* __restrict__ mx) {
  __shared__ float r0[256], r1[256], r2[256];
  int gid = blockIdx.x * 256 + threadIdx.x;
  float s = 0.0f, dh = 0.0f, dp = 0.0f;
  if (gid < E_ELEMS) {
    float S  = mel[gid];
    float ha = harm[gid];
    float pe = perc[gid];
    float Z  = fmaxf(ha, pe);
    bool bad = Z < 1.17549435e-38f;          // float32 tiny
    float Zs = bad ? 1.0f : Z;
    float mh = ha / Zs; mh *= mh;
    float mp = pe / Zs; mp *= mp;
    float den = mh + mp;
    float maskh = bad ? 0.0f : mh / den;
    float maskp = bad ? 0.0f : mp / den;
    dh = S * maskh; dp = S * maskp; s = S;
    harm[gid] = dh;                          // overwrite in place: D_h, D_p
    perc[gid] = dp;
  }
  int tid = threadIdx.x;
  r0[tid] = s; r1[tid] = dh; r2[tid] = dp;
  __syncthreads();
  for (int off = 128; off > 0; off >>= 1) {
    if (tid < off) {
      r0[tid] = fmaxf(r0[tid], r0[tid + off]);
      r1[tid] = fmaxf(r1[tid], r1[tid + off]);
      r2[tid] = fmaxf(r2[tid], r2[tid + off]);
    }
    __syncthreads();
  }
  if (tid == 0) {                            // all values >= 0: uint-bit max
    atomicMax(mx + 0, __float_as_uint(r0[0]));
    atomicMax(mx + 1, __float_as_uint(r1[0]));
    atomicMax(mx + 2, __float_as_uint(r2[0]));
  }
}

// ============ kernel 6: power_to_db + normalize + concat ====================
__global__ __launch_bounds__(256) void finalize(const float* __restrict__ mel,
                                                const float* __restrict__ harm,
                                                const float* __restrict__ perc,
                                                const unsigned* __restrict__ mx,
                                                float* __restrict__ out) {
  int gid = blockIdx.x * 256 + threadIdx.x;
  if (gid >= 3 * E_ELEMS) return;
  int s = gid / E_ELEMS;
  int e = gid - s * E_ELEMS;
  const float* src = (s == 0) ? mel : ((s == 1) ? harm : perc);
  float ref = __uint_as_float(mx[s]);
  float v   = src[e];
  // max(log_spec) == 0 always, so top_db clamp is exactly -80 dB
  float db = 10.0f * log10f(fmaxf(1e-10f, v)) -
             10.0f * log10f(fmaxf(1e-10f, ref));
  db = fmaxf(db, -80.0f);
  out[gid] = (db + 80.0f) * (1.0f / (80.0f + 1e-7f));
}

// ================= host launcher =================
extern "C" void kernel_launch(void* const* d_in, const int* in_sizes, int n_in,
                              void* d_out, int out_size, void* d_ws,
                              size_t ws_size, hipStream_t stream) {
  const float* wave = (const float*)d_in[0];
  float* out  = (float*)d_out;
  float* ws   = (float*)d_ws;
  float* mel  = ws + MEL_OFF;
  float* harm = ws + HARM_OFF;
  float* perc = ws + PERC_OFF;
  unsigned* mx = (unsigned*)(ws + MAX_OFF);

  build_tables<<<MAX_OFF / 256, 256, 0, stream>>>(ws);
  mel_gemm<<<NB * MTILES, 256, 0, stream>>>(wave, ws, mel);
  int mblocks = (E_ELEMS + 255) / 256;
  median_time<<<mblocks, 256, 0, stream>>>(mel, harm);
  median_mel<<<mblocks, 256, 0, stream>>>(mel, perc);
  hpss_mask<<<mblocks, 256, 0, stream>>>(mel, harm, perc, mx);
  finalize<<<(3 * E_ELEMS + 255) / 256, 256, 0, stream>>>(mel, harm, perc, mx,
                                                          out);
}